// ProductLoss_51367808860812
// MI455X (gfx1250) — compile-verified
//
#include <hip/hip_runtime.h>
#include <hip/hip_bf16.h>

typedef __attribute__((ext_vector_type(2))) float v2f;
typedef __attribute__((ext_vector_type(8))) float v8f;

// -----------------------------------------------------------------------------
// Kernel 1: head GEMM via V_WMMA_F32_16X16X4_F32 (exact f32 path).
//   G[b][n]  (256 x 1024 f32, row-major) where
//     n <  512 : G = E @ W1[0:512, :]                  ("A" half, indexed by j)
//     n >= 512 : G = E @ W1[512:1024, :] + b1          ("C" half, indexed by i)
//   E:  256 x 512 row-major;  W1: 1024 x 512 row-major (d-major).
// One 16x16 output tile per wave, K swept in steps of 4 (128 WMMAs/tile).
// Fragment layouts per CDNA5 ISA 7.12.2:
//   A (16x4 f32): lane m = lane&15; v0 = K(2*(lane>>4)), v1 = K(2*(lane>>4)+1)
//   B (4x16 f32): lane n = lane&15; mirrored K split across lane halves
//   C/D         : VGPR r -> M = r + 8*(lane>>4), N = lane&15
// -----------------------------------------------------------------------------
__global__ __launch_bounds__(256) void head_gemm_wmma(
    const float* __restrict__ E, const float* __restrict__ W1,
    const float* __restrict__ b1, float* __restrict__ G)
{
    const int lane  = threadIdx.x & 31;
    const int wave  = threadIdx.x >> 5;
    const int tile  = blockIdx.x * 8 + wave;   // 0..1023 (16 m-tiles x 64 n-tiles)
    const int nt    = tile & 63;
    const int mt    = tile >> 6;
    const int l16   = lane & 15;
    const int khalf = lane >> 4;               // 0 or 1

    const int m    = mt * 16 + l16;            // A-fragment row (M)
    const int ncol = nt * 16 + l16;            // B-fragment col (N), 0..1023

    const float* __restrict__ erow = E + (size_t)m * 512;
    // Column base into W1 for this output column (top vs bottom weight half).
    const float* __restrict__ wcol =
        (ncol < 512) ? (W1 + ncol) : (W1 + (size_t)512 * 512 + (ncol - 512));

    v8f acc = {};
    #pragma unroll 4
    for (int k0 = 0; k0 < 512; k0 += 4) {
        const int ka = k0 + 2 * khalf;
        // A fragment: two consecutive K of row m (8B-aligned -> one b64 load)
        v2f a = *(const v2f*)(erow + ka);
        // B fragment: two K-rows of column ncol (stride-H, scalar loads)
        v2f b;
        b.x = wcol[(size_t)ka * 512];
        b.y = wcol[(size_t)(ka + 1) * 512];
        acc = __builtin_amdgcn_wmma_f32_16x16x4_f32(
            /*neg_a=*/false, a, /*neg_b=*/false, b,
            /*c_mod=*/(short)0, acc, /*reuse_a=*/false, /*reuse_b=*/false);
    }

    const float bias = (ncol >= 512) ? b1[ncol - 512] : 0.0f;  // fold b1 into C half
    #pragma unroll
    for (int r = 0; r < 8; ++r) {
        const int mr = mt * 16 + r + 8 * khalf;
        G[(size_t)mr * 1024 + ncol] = acc[r] + bias;
    }
}

// -----------------------------------------------------------------------------
// Kernel 2: pair loss.
//   out[i*256 + j] = BCE( labels[i]==labels[j],
//                         clip(sigmoid( sum_h relu(G[j][h] + G[i][512+h]) * W2[h] + b2 )) )
// 64 WGs x 256 threads; 32x32 (i,j) tile per WG, 2x2 microtile per thread,
// H staged through LDS in 64-wide chunks (rows padded to 68 floats: 272B = 17*16B).
// -----------------------------------------------------------------------------
__global__ __launch_bounds__(256) void pair_loss_kernel(
    const float* __restrict__ G, const float* __restrict__ W2,
    const float* __restrict__ b2, const int* __restrict__ labels,
    float* __restrict__ out)
{
    __shared__ float As[32 * 68];   // A rows (j-indexed), padded stride 68
    __shared__ float Cs[32 * 68];   // C rows (i-indexed), padded stride 68
    __shared__ float Ws[64];        // W2 chunk

    const int t  = threadIdx.x;
    const int bi = blockIdx.x >> 3;        // 8x8 grid of 32x32 tiles
    const int bj = blockIdx.x & 7;
    const int i0 = bi * 32, j0 = bj * 32;
    const int tj = t & 15, ti = t >> 4;

    float acc00 = 0.f, acc01 = 0.f, acc10 = 0.f, acc11 = 0.f;

    for (int h0 = 0; h0 < 512; h0 += 64) {
        #pragma unroll
        for (int r = 0; r < 2; ++r) {
            const int idx = t + 256 * r;       // 0..511 -> 32 rows x 16 float4
            const int row = idx >> 4;
            const int c4  = (idx & 15) * 4;
            *(float4*)&As[row * 68 + c4] =
                *(const float4*)&G[(size_t)(j0 + row) * 1024 + h0 + c4];
            *(float4*)&Cs[row * 68 + c4] =
                *(const float4*)&G[(size_t)(i0 + row) * 1024 + 512 + h0 + c4];
        }
        if (t < 64) Ws[t] = W2[h0 + t];
        __syncthreads();

        #pragma unroll 8
        for (int hh = 0; hh < 64; ++hh) {
            const float w  = Ws[hh];
            const float a0 = As[(tj     ) * 68 + hh];
            const float a1 = As[(tj + 16) * 68 + hh];
            const float c0 = Cs[(ti     ) * 68 + hh];
            const float c1 = Cs[(ti + 16) * 68 + hh];
            acc00 += fmaxf(c0 + a0, 0.f) * w;
            acc01 += fmaxf(c0 + a1, 0.f) * w;
            acc10 += fmaxf(c1 + a0, 0.f) * w;
            acc11 += fmaxf(c1 + a1, 0.f) * w;
        }
        __syncthreads();
    }

    const float bb = b2[0];
    float accs[2][2] = {{acc00, acc01}, {acc10, acc11}};
    #pragma unroll
    for (int ii = 0; ii < 2; ++ii) {
        const int i  = i0 + ti + 16 * ii;
        const int li = labels[i];
        #pragma unroll
        for (int jj = 0; jj < 2; ++jj) {
            const int j = j0 + tj + 16 * jj;
            const float z = accs[ii][jj] + bb;
            float p = 1.0f / (1.0f + __expf(-z));
            p = fminf(fmaxf(p, 1e-7f), 1.0f - 1e-7f);
            const bool same = (li == labels[j]);
            const float loss = same ? -__logf(p) : -__logf(1.0f - p);
            out[(size_t)i * 256 + j] = loss;
        }
    }
}

extern "C" void kernel_launch(void* const* d_in, const int* in_sizes, int n_in,
                              void* d_out, int out_size, void* d_ws, size_t ws_size,
                              hipStream_t stream) {
    const float* E      = (const float*)d_in[0];   // (256, 512)
    const int*   labels = (const int*)  d_in[1];   // (256,)
    const float* W1     = (const float*)d_in[2];   // (1024, 512)
    const float* b1     = (const float*)d_in[3];   // (512,)
    const float* W2     = (const float*)d_in[4];   // (512, 1)
    const float* b2     = (const float*)d_in[5];   // (1,)
    float*       out    = (float*)d_out;           // (65536,)
    float*       G      = (float*)d_ws;            // 256*1024 f32 = 1 MB scratch

    // 1024 output tiles, 8 waves per 256-thread block -> 128 blocks.
    head_gemm_wmma<<<128, 256, 0, stream>>>(E, W1, b1, G);
    // 8x8 grid of 32x32 (i,j) tiles -> 64 blocks.
    pair_loss_kernel<<<64, 256, 0, stream>>>(G, W2, b2, labels, out);
}